// LeftPool_61452392071363
// MI455X (gfx1250) — compile-verified
//
#include <hip/hip_runtime.h>
#include <math.h>

// LeftPool: out[b,c,h,w] = max(x[b,c,h,w:])  -- reverse cummax over W=128.
// Shape (16,256,128,128) fp32. Memory-bound: 512 MB traffic, ~22 us floor @ 23.3 TB/s.
// Strategy: one wave32 per row; float4 per lane; wave suffix-max scan via shuffles;
// non-temporal 128-bit loads/stores (streaming data, touched once -> don't pollute L2).

typedef __attribute__((ext_vector_type(4))) float v4f;

#define ROW_W 128               // floats per row
#define LANES 32                // wave32: 32 lanes * 4 floats = 128

__global__ __launch_bounds__(256)
void leftpool_rcummax_kernel(const float* __restrict__ x,
                             float* __restrict__ out,
                             int nrows) {
  const int lane         = threadIdx.x & (LANES - 1);
  const int waveInBlock  = threadIdx.x >> 5;
  const int wavesPerBlk  = blockDim.x >> 5;
  const int row          = blockIdx.x * wavesPerBlk + waveInBlock;
  if (row >= nrows) return;

  const size_t rowBase = (size_t)row * ROW_W;

  // One coalesced 512B burst per wave: lane i holds elements [4i, 4i+3].
  const v4f* src = reinterpret_cast<const v4f*>(x + rowBase) + lane;
  v4f v = __builtin_nontemporal_load(src);   // global_load_b128, TH=NT

  // Lane-local reverse (suffix) max over the 4 elements; m0 = lane segment max.
  float m3 = v.w;
  float m2 = fmaxf(v.z, m3);
  float m1 = fmaxf(v.y, m2);
  float m0 = fmaxf(v.x, m1);

  // Inclusive suffix-max scan across the 32 lanes (5 shuffle-down steps).
  float s = m0;
  #pragma unroll
  for (int off = 1; off < LANES; off <<= 1) {
    float o = __shfl_down(s, off, LANES);
    s = (lane + off < LANES) ? fmaxf(s, o) : s;
  }

  // Exclusive suffix: max over lanes strictly to the right; identity = -inf.
  float e = __shfl_down(s, 1, LANES);
  if (lane == LANES - 1) e = -INFINITY;

  // Fold the cross-lane suffix into the lane-local reverse cummax.
  v4f r;
  r.w = fmaxf(v.w, e);
  r.z = fmaxf(v.z, r.w);
  r.y = fmaxf(v.y, r.z);
  r.x = fmaxf(v.x, r.y);

  v4f* dst = reinterpret_cast<v4f*>(out + rowBase) + lane;
  __builtin_nontemporal_store(r, dst);       // global_store_b128, TH=NT
}

extern "C" void kernel_launch(void* const* d_in, const int* in_sizes, int n_in,
                              void* d_out, int out_size, void* d_ws, size_t ws_size,
                              hipStream_t stream) {
  (void)in_sizes; (void)n_in; (void)d_ws; (void)ws_size;

  const float* x  = (const float*)d_in[0];
  float*       o  = (float*)d_out;

  // rows = total elements / W. For the reference shape: 16*256*128 = 524288 rows.
  const int nrows = out_size / ROW_W;

  const int threads       = 256;            // 8 waves per block (wave32)
  const int wavesPerBlock = threads / 32;
  const int blocks        = (nrows + wavesPerBlock - 1) / wavesPerBlock;

  leftpool_rcummax_kernel<<<blocks, threads, 0, stream>>>(x, o, nrows);
}